// SortingRegressionModel_35785667510837
// MI455X (gfx1250) — compile-verified
//
#include <hip/hip_runtime.h>

// out[r] = dot(sort(x[r, 0:3]), W[0, 0:3]) + b[0]   for r in [0, B)
//
// Memory-bound streaming kernel: 8 rows per thread so the per-thread input
// (24 floats = 96 B) is exactly six global_load_b128 and the output
// (8 floats = 32 B) is two global_store_b128. Sort-3 is done with the
// min3 / med3 / max3 single-instruction network; the dot is an FMA chain.

#define TPB   256   // 8 wave32 waves per block
#define RPT   8     // rows per thread

__global__ __launch_bounds__(TPB)
void sort3_linear_kernel(const float* __restrict__ x,
                         const float* __restrict__ W,
                         const float* __restrict__ b,
                         float* __restrict__ out,
                         int nrows)
{
    const long long tid  = (long long)blockIdx.x * TPB + threadIdx.x;
    const long long row0 = tid * RPT;
    if (row0 >= nrows) return;

    // Uniform across the grid -> compiler emits scalar loads (s_load_b32).
    const float w0   = W[0];
    const float w1   = W[1];
    const float w2   = W[2];
    const float bias = b[0];

    if (row0 + RPT <= nrows) {
        // Fast path: 6 x b128 loads (96 contiguous bytes per thread).
        float4 v[6];
        const float4* xv = reinterpret_cast<const float4*>(x + row0 * 3);
        #pragma unroll
        for (int i = 0; i < 6; ++i) v[i] = xv[i];

        const float* f = reinterpret_cast<const float*>(v);
        float res[RPT];
        #pragma unroll
        for (int r = 0; r < RPT; ++r) {
            const float a = f[r * 3 + 0];
            const float c = f[r * 3 + 1];
            const float e = f[r * 3 + 2];
            // sort3: v_min3_f32 / v_med3_f32 / v_max3_f32
            const float s0 = fminf(fminf(a, c), e);
            const float s1 = __builtin_amdgcn_fmed3f(a, c, e);
            const float s2 = fmaxf(fmaxf(a, c), e);
            res[r] = fmaf(s0, w0, fmaf(s1, w1, fmaf(s2, w2, bias)));
        }

        // 2 x b128 stores (32 contiguous bytes per thread).
        float4* ov = reinterpret_cast<float4*>(out + row0);
        ov[0] = make_float4(res[0], res[1], res[2], res[3]);
        ov[1] = make_float4(res[4], res[5], res[6], res[7]);
    } else {
        // Tail path (not hit for B = 8388608, kept for generality).
        for (long long r = row0; r < nrows; ++r) {
            const float a = x[r * 3 + 0];
            const float c = x[r * 3 + 1];
            const float e = x[r * 3 + 2];
            const float s0 = fminf(fminf(a, c), e);
            const float s1 = __builtin_amdgcn_fmed3f(a, c, e);
            const float s2 = fmaxf(fmaxf(a, c), e);
            out[r] = fmaf(s0, w0, fmaf(s1, w1, fmaf(s2, w2, bias)));
        }
    }
}

extern "C" void kernel_launch(void* const* d_in, const int* in_sizes, int n_in,
                              void* d_out, int out_size, void* d_ws, size_t ws_size,
                              hipStream_t stream)
{
    const float* x = (const float*)d_in[0];   // [B, 3] f32
    const float* W = (const float*)d_in[1];   // [1, 3] f32
    const float* b = (const float*)d_in[2];   // [1]    f32
    float* out     = (float*)d_out;           // [B, 1] f32

    const int nrows = in_sizes[0] / 3;        // 8388608
    const long long rows_per_block = (long long)TPB * RPT;       // 2048
    const int grid = (int)((nrows + rows_per_block - 1) / rows_per_block); // 4096

    sort3_linear_kernel<<<grid, TPB, 0, stream>>>(x, W, b, out, nrows);
}